// MattNet_15719580303465
// MI455X (gfx1250) — compile-verified
//
#include <hip/hip_runtime.h>
#include <hip/hip_bf16.h>
#include <math.h>

// CDNA5 (gfx1250) wave32 WMMA types
typedef __attribute__((ext_vector_type(2))) float v2f;
typedef __attribute__((ext_vector_type(8))) float v8f;

#define KIMG 27

// One wave (32 lanes) owns a 16x16 (i,j) tile of one structure b and loops over
// the 27 periodic images. Squared distances come from V_WMMA_F32_16X16X4_F32:
//   D = A(16x4) x B(4x16) + C
//   A row M  = [-2*pix, -2*piy, -2*piz, 1]
//   B col N  = [ qx, qy, qz, |q|^2 ],  q = pos_j + offset_cart[k]
//   C[M][N]  = |p_i(M)|^2
// => D[M][N] = |p_i|^2 - 2 p_i.q + |q|^2 = squared distance (exact fp32 FMA).
__global__ __launch_bounds__(256) void pbc_tile_kernel(
    const float* __restrict__ pos,   // [B,N,3]
    const float* __restrict__ cell,  // [B,3,3] row vectors
    float* __restrict__ dist,        // [B,N,N,K]
    float* __restrict__ dvec,        // [B,N,N,K,3]
    float* __restrict__ maskf,       // [B,N,N,K] (0.0 / 1.0)
    int B, int N, int totalTiles)
{
    const int lane = threadIdx.x & 31;
    const int wave = threadIdx.x >> 5;
    const int tile = blockIdx.x * 8 + wave;
    if (tile >= totalTiles) return;   // wave-uniform exit: EXEC stays all-1s

    const int nt  = N >> 4;
    const int b   = tile / (nt * nt);
    const int rem = tile - b * (nt * nt);
    const int i0  = (rem / nt) << 4;
    const int j0  = (rem % nt) << 4;

    const float* pb = pos + (size_t)b * N * 3;
    const float* cb = cell + (size_t)b * 9;

    __builtin_prefetch(pb, 0, 0);     // global_prefetch_b8: pull pos into cache

    const bool hi = (lane >= 16);
    const int  n  = lane & 15;        // column N of B and of C/D for this lane
    const int  j  = j0 + n;

    // cell row vectors (uniform per wave)
    const float c00 = cb[0], c01 = cb[1], c02 = cb[2];
    const float c10 = cb[3], c11 = cb[4], c12 = cb[5];
    const float c20 = cb[6], c21 = cb[7], c22 = cb[8];

    // ---- A operand: 16x4 f32. Lanes 0-15: M=lane, VGPR0/1 = K0/K1;
    //                             lanes 16-31: M=lane-16, VGPR0/1 = K2/K3.
    const float amx = pb[(i0 + n) * 3 + 0];
    const float amy = pb[(i0 + n) * 3 + 1];
    const float amz = pb[(i0 + n) * 3 + 2];
    v2f a;
    a.x = hi ? (-2.0f * amz) : (-2.0f * amx);
    a.y = hi ? 1.0f          : (-2.0f * amy);

    // ---- pos_j column for this lane (both halves hold column n)
    const float pjx = pb[j * 3 + 0];
    const float pjy = pb[j * 3 + 1];
    const float pjz = pb[j * 3 + 2];

    // ---- C operand + per-lane row data. C/D VGPR r: M = r + (hi ? 8 : 0).
    // Also hoist the per-row output base index (32-bit: max 5*S1*? fits u32).
    float pix[8], piy[8], piz[8];
    unsigned ebase[8];     // element index of (b,i,j,k=0) in [B,N,N,K]
    bool     diag[8];      // i == j (self-pair candidate)
    v8f cacc;
    const unsigned bOff = (unsigned)b * (unsigned)N * (unsigned)N * KIMG;
    #pragma unroll
    for (int r = 0; r < 8; ++r) {
        const int i = i0 + r + (hi ? 8 : 0);
        pix[r] = pb[i * 3 + 0];
        piy[r] = pb[i * 3 + 1];
        piz[r] = pb[i * 3 + 2];
        cacc[r] = pix[r] * pix[r] + piy[r] * piy[r] + piz[r] * piz[r];
        ebase[r] = bOff + ((unsigned)i * (unsigned)N + (unsigned)j) * KIMG;
        diag[r] = (i == j);
    }

    for (int k = 0; k < KIMG; ++k) {
        // frac[k] = (k/9 - 1, (k/3)%3 - 1, k%3 - 1); offset = frac @ cell
        const float fx = (float)(k / 9 - 1);
        const float fy = (float)((k / 3) % 3 - 1);
        const float fz = (float)(k % 3 - 1);
        const float ox = fx * c00 + fy * c10 + fz * c20;
        const float oy = fx * c01 + fy * c11 + fz * c21;
        const float oz = fx * c02 + fy * c12 + fz * c22;

        const float qx = pjx + ox, qy = pjy + oy, qz = pjz + oz;
        const float q2 = qx * qx + qy * qy + qz * qz;
        const bool  kzero = (k == 13);

        // ---- B operand: 4x16 f32. Lanes 0-15: VGPR0/1 = row K0/K1 (qx/qy);
        //                            lanes 16-31: VGPR0/1 = row K2/K3 (qz/|q|^2).
        v2f bv;
        bv.x = hi ? qz : qx;
        bv.y = hi ? q2 : qy;

        // D = A x B + C  (8 args: neg_a, A, neg_b, B, c_mod, C, reuse_a, reuse_b)
        v8f d = __builtin_amdgcn_wmma_f32_16x16x4_f32(
            false, a, false, bv, (short)0, cacc, false, false);

        #pragma unroll
        for (int r = 0; r < 8; ++r) {
            const float sq = d[r];
            // reference mask: EPS < sq <= cutoff^2, excluding the exact self
            // pair (zero image), which the expanded bilinear form can't
            // reproduce as an exact 0.
            const bool mk = (sq > 1e-8f) && (sq <= 25.0f) && !(diag[r] && kzero);
            const float m = mk ? 1.0f : 0.0f;

            // branch-free safe sqrt: sqrt(where(mask, sq, 1.0)) * m
            const float ssel = mk ? sq : 1.0f;             // v_cndmask
            const float dv   = __builtin_amdgcn_sqrtf(ssel) * m;  // v_sqrt_f32

            const unsigned e = ebase[r] + (unsigned)k;
            dist[e]  = dv;
            maskf[e] = m;

            float* vp = dvec + (size_t)e * 3u;
            vp[0] = (qx - pix[r]) * m;
            vp[1] = (qy - piy[r]) * m;
            vp[2] = (qz - piz[r]) * m;
        }
    }
}

__global__ void offset_cart_kernel(const float* __restrict__ cell,
                                   float* __restrict__ offc, int B)
{
    const int b = blockIdx.x;
    const int k = threadIdx.x;
    if (k < KIMG) {
        const float fx = (float)(k / 9 - 1);
        const float fy = (float)((k / 3) % 3 - 1);
        const float fz = (float)(k % 3 - 1);
        const float* cb = cell + (size_t)b * 9;
        float* o = offc + ((size_t)b * KIMG + k) * 3;
        o[0] = fx * cb[0] + fy * cb[3] + fz * cb[6];
        o[1] = fx * cb[1] + fy * cb[4] + fz * cb[7];
        o[2] = fx * cb[2] + fy * cb[5] + fz * cb[8];
    }
}

extern "C" void kernel_launch(void* const* d_in, const int* in_sizes, int n_in,
                              void* d_out, int out_size, void* d_ws, size_t ws_size,
                              hipStream_t stream)
{
    (void)n_in; (void)out_size; (void)d_ws; (void)ws_size;

    const float* pos  = (const float*)d_in[0];  // [B,N,3]
    const float* cell = (const float*)d_in[1];  // [B,3,3]

    const int B = in_sizes[1] / 9;
    const int N = in_sizes[0] / (3 * B);

    const size_t S1 = (size_t)B * N * N * KIMG;
    float* out   = (float*)d_out;
    float* dist  = out;            // [B,N,N,K]
    float* dvec  = out + S1;       // [B,N,N,K,3]
    float* maskf = out + 4 * S1;   // [B,N,N,K]
    float* offc  = out + 5 * S1;   // [B,K,3]

    const int nt = N / 16;
    const int totalTiles = B * nt * nt;          // 4096 waves for B=4, N=512
    const int blocks = (totalTiles + 7) / 8;     // 8 waves (256 threads) per block

    pbc_tile_kernel<<<blocks, 256, 0, stream>>>(pos, cell, dist, dvec, maskf,
                                                B, N, totalTiles);
    offset_cart_kernel<<<B, 32, 0, stream>>>(cell, offc, B);
}